// Seq2Seq_39676907888557
// MI455X (gfx1250) — compile-verified
//
#include <hip/hip_runtime.h>
#include <cstddef>

// ---------------- problem constants ----------------
#define Bsz   16
#define Lsz   400
#define Esz   128
#define Hsz   256
#define H2sz  512
#define Vsz   50000
#define Tsz   64
#define VEXT  50050   // V + MAX_OOV(50)

typedef __attribute__((ext_vector_type(2))) float v2f;
typedef __attribute__((ext_vector_type(8))) float v8f;
typedef __attribute__((ext_vector_type(4))) unsigned int v4u;
typedef __attribute__((ext_vector_type(4))) int v4i;
typedef __attribute__((ext_vector_type(8))) int v8i;

__device__ __forceinline__ float sigf(float x) { return 1.0f / (1.0f + __expf(-x)); }

// WMMA fp32 16x16x4:  D = A(16x4) * B(4x16) + C
#define WMMA4(a, b, c) __builtin_amdgcn_wmma_f32_16x16x4_f32(false, (a), false, (b), (short)0, (c), false, false)

#if defined(__HIP_DEVICE_COMPILE__) && __has_builtin(__builtin_amdgcn_tensor_load_to_lds)
#define HAVE_TDM 1
#else
#define HAVE_TDM 0
#endif

// ---------------- init: tsum=0, dec_in=START_ID ----------------
__global__ __launch_bounds__(256) void init_kernel(float* tsum, int* dec_in) {
  int tid = threadIdx.x;
  for (int i = tid; i < Bsz * Lsz; i += 256) tsum[i] = 0.0f;
  if (tid < Bsz) dec_in[tid] = 1;  // START_ID
}

// ---------------- masked embedding gather: emb[B][L][E] ----------------
__global__ __launch_bounds__(128) void embed_kernel(const int* __restrict__ x,
                                                    const int* __restrict__ x_len,
                                                    const float* __restrict__ embedding,
                                                    float* __restrict__ emb) {
  int bl = blockIdx.x;              // 0..B*L-1
  int b = bl / Lsz, l = bl % Lsz;
  int e = threadIdx.x;
  int id = x[bl];
  float v = embedding[(size_t)id * Esz + e];
  emb[(size_t)bl * Esz + e] = (l < x_len[b]) ? v : 0.0f;
}

// ---------------- bidirectional encoder LSTM (persistent, 400 steps) ----------------
__global__ __launch_bounds__(512) void enc_lstm_kernel(
    const float* __restrict__ emb, const int* __restrict__ x_len,
    const float* __restrict__ Wx_f, const float* __restrict__ Wh_f, const float* __restrict__ b_f,
    const float* __restrict__ Wx_b, const float* __restrict__ Wh_b, const float* __restrict__ b_b,
    float* __restrict__ enc, float* __restrict__ h0, float* __restrict__ c0) {
  const int dir = blockIdx.x;
  const float* Wx = dir ? Wx_b : Wx_f;
  const float* Wh = dir ? Wh_b : Wh_f;
  const float* bs = dir ? b_b : b_f;
  const int tid = threadIdx.x, wave = tid >> 5, lane = tid & 31;
  const int m15 = lane & 15, kh = lane >> 4;
  const int col = wave * 16 + m15;          // output H column

  __shared__ float h_s[Bsz][Hsz + 8];
  __shared__ int xl_s[Bsz];
  for (int i = tid; i < Bsz * (Hsz + 8); i += 512) ((float*)h_s)[i] = 0.0f;
  if (tid < Bsz) xl_s[tid] = x_len[tid];

  v8f c_reg, h_reg;
  for (int r = 0; r < 8; ++r) { c_reg[r] = 0.0f; h_reg[r] = 0.0f; }
  __syncthreads();

  for (int step = 0; step < Lsz; ++step) {
    const int t = dir ? (Lsz - 1 - step) : step;
    v8f acc[4];
    for (int g = 0; g < 4; ++g) {
      float bb = bs[g * Hsz + col];
      for (int r = 0; r < 8; ++r) acc[g][r] = bb;
    }
    // x_t @ Wx  (K = 128)
    const float* xrow = emb + ((size_t)m15 * Lsz + t) * Esz;
#pragma unroll 4
    for (int k = 0; k < Esz; k += 4) {
      int kk = k + 2 * kh;
      v2f a; a.x = xrow[kk]; a.y = xrow[kk + 1];
      for (int g = 0; g < 4; ++g) {
        int n = g * Hsz + col;
        v2f bm; bm.x = Wx[(size_t)kk * 1024 + n]; bm.y = Wx[(size_t)(kk + 1) * 1024 + n];
        acc[g] = WMMA4(a, bm, acc[g]);
      }
    }
    // h @ Wh  (K = 256)
#pragma unroll 4
    for (int k = 0; k < Hsz; k += 4) {
      int kk = k + 2 * kh;
      v2f a; a.x = h_s[m15][kk]; a.y = h_s[m15][kk + 1];
      for (int g = 0; g < 4; ++g) {
        int n = g * Hsz + col;
        v2f bm; bm.x = Wh[(size_t)kk * 1024 + n]; bm.y = Wh[(size_t)(kk + 1) * 1024 + n];
        acc[g] = WMMA4(a, bm, acc[g]);
      }
    }
    // LSTM cell + state freeze at padded positions
    for (int r = 0; r < 8; ++r) {
      int mr = r + 8 * kh;
      float cn = sigf(acc[1][r]) * c_reg[r] + sigf(acc[0][r]) * tanhf(acc[2][r]);
      float hn = sigf(acc[3][r]) * tanhf(cn);
      if (t < xl_s[mr]) { c_reg[r] = cn; h_reg[r] = hn; }
      enc[((size_t)mr * Lsz + t) * H2sz + dir * Hsz + col] = h_reg[r];
    }
    __syncthreads();
    for (int r = 0; r < 8; ++r) h_s[r + 8 * kh][col] = h_reg[r];
    __syncthreads();
  }
  for (int r = 0; r < 8; ++r) {
    int mr = r + 8 * kh;
    h0[mr * H2sz + dir * Hsz + col] = h_reg[r];
    c0[mr * H2sz + dir * Hsz + col] = c_reg[r];
  }
}

// ---------------- decoder LSTM cell (one step i), 32 blocks x 1 wave ----------------
__global__ __launch_bounds__(32) void dec_cell_kernel(
    const float* __restrict__ embedding, const int* __restrict__ dec_in,
    const float* __restrict__ Wx_d, const float* __restrict__ Wh_d, const float* __restrict__ b_d,
    const float* __restrict__ h0, const float* __restrict__ c_in,
    float* __restrict__ pre_h, float* __restrict__ c_out, int i) {
  const int lane = threadIdx.x & 31;
  const int m15 = lane & 15, kh = lane >> 4;
  const int col = blockIdx.x * 16 + m15;     // 0..511

  int tok = dec_in[m15];
  if (tok >= Vsz) tok = Vsz - 1;             // JAX OOB gather clamps
  const float* xrow = embedding + (size_t)tok * Esz;
  const float* hrow = (i == 0) ? (h0 + m15 * H2sz)
                               : (pre_h + ((size_t)m15 * Tsz + (i - 1)) * H2sz);
  v8f acc[4];
  for (int g = 0; g < 4; ++g) {
    float bb = b_d[g * H2sz + col];
    for (int r = 0; r < 8; ++r) acc[g][r] = bb;
  }
#pragma unroll 4
  for (int k = 0; k < Esz; k += 4) {
    int kk = k + 2 * kh;
    v2f a; a.x = xrow[kk]; a.y = xrow[kk + 1];
    for (int g = 0; g < 4; ++g) {
      int n = g * H2sz + col;
      v2f bm; bm.x = Wx_d[(size_t)kk * 2048 + n]; bm.y = Wx_d[(size_t)(kk + 1) * 2048 + n];
      acc[g] = WMMA4(a, bm, acc[g]);
    }
  }
#pragma unroll 2
  for (int k = 0; k < H2sz; k += 4) {
    int kk = k + 2 * kh;
    v2f a; a.x = hrow[kk]; a.y = hrow[kk + 1];
    for (int g = 0; g < 4; ++g) {
      int n = g * H2sz + col;
      v2f bm; bm.x = Wh_d[(size_t)kk * 2048 + n]; bm.y = Wh_d[(size_t)(kk + 1) * 2048 + n];
      acc[g] = WMMA4(a, bm, acc[g]);
    }
  }
  for (int r = 0; r < 8; ++r) {
    int mr = r + 8 * kh;
    float co = c_in[mr * H2sz + col];
    float cn = sigf(acc[1][r]) * co + sigf(acc[0][r]) * tanhf(acc[2][r]);
    float hn = sigf(acc[3][r]) * tanhf(cn);
    c_out[mr * H2sz + col] = cn;
    pre_h[((size_t)mr * Tsz + i) * H2sz + col] = hn;
  }
}

// ---------------- q = h2 @ W_enc ; qd = h2 @ W_dec  (64 blocks x 1 wave) ----------------
__global__ __launch_bounds__(32) void attn_q_kernel(const float* __restrict__ pre_h,
                                                    const float* __restrict__ W_enc,
                                                    const float* __restrict__ W_dec,
                                                    float* __restrict__ q, float* __restrict__ qd, int i) {
  const int lane = threadIdx.x & 31;
  const int m15 = lane & 15, kh = lane >> 4;
  const int mat = blockIdx.x >> 5;           // 0 -> W_enc, 1 -> W_dec
  const int blk = blockIdx.x & 31;
  const int col = blk * 16 + m15;
  const float* W = mat ? W_dec : W_enc;
  float* outp = mat ? qd : q;
  const float* arow = pre_h + ((size_t)m15 * Tsz + i) * H2sz;
  v8f acc; for (int r = 0; r < 8; ++r) acc[r] = 0.0f;
#pragma unroll 2
  for (int k = 0; k < H2sz; k += 4) {
    int kk = k + 2 * kh;
    v2f a; a.x = arow[kk]; a.y = arow[kk + 1];
    v2f bm; bm.x = W[(size_t)kk * H2sz + col]; bm.y = W[(size_t)(kk + 1) * H2sz + col];
    acc = WMMA4(a, bm, acc);
  }
  for (int r = 0; r < 8; ++r) outp[(r + 8 * kh) * H2sz + col] = acc[r];
}

// ---------------- attention scores: exp_e[b][l], raw d[b][t] ----------------
__global__ __launch_bounds__(512) void attn_scores_kernel(
    const float* __restrict__ q, const float* __restrict__ qd,
    const float* __restrict__ enc, const float* __restrict__ pre_h,
    const int* __restrict__ x_len, float* __restrict__ expe, float* __restrict__ dvals) {
  const int b = blockIdx.x;
  const int wave = threadIdx.x >> 5, lane = threadIdx.x & 31;
  const float* qb = q + b * H2sz;
  const int xl = x_len[b];
  for (int li = 0; li < 25; ++li) {
    int l = wave * 25 + li;
    const float* er = enc + ((size_t)b * Lsz + l) * H2sz;
    float s = 0.0f;
    for (int j = lane; j < H2sz; j += 32) s += qb[j] * er[j];
    for (int off = 16; off > 0; off >>= 1) s += __shfl_xor(s, off, 32);
    if (lane == 0) expe[b * Lsz + l] = (l < xl) ? __expf(s) : 0.0f;
  }
  const float* qdb = qd + b * H2sz;
  for (int tt = 0; tt < 4; ++tt) {
    int t = wave * 4 + tt;
    const float* pr = pre_h + ((size_t)b * Tsz + t) * H2sz;
    float s = 0.0f;
    for (int j = lane; j < H2sz; j += 32) s += qdb[j] * pr[j];
    for (int off = 16; off > 0; off >>= 1) s += __shfl_xor(s, off, 32);
    if (lane == 0) dvals[b * Tsz + t] = s;
  }
}

// ---------------- temporal attention, contexts, comb, p ----------------
__global__ __launch_bounds__(512) void attn_finalize_kernel(
    float* __restrict__ expe, float* __restrict__ tsum,
    const float* __restrict__ enc, const float* __restrict__ dvals,
    const float* __restrict__ pre_h, const float* __restrict__ Wp, const float* __restrict__ bp,
    float* __restrict__ att, float* __restrict__ comb, float* __restrict__ pval, int i) {
  const int b = blockIdx.x, tid = threadIdx.x;
  __shared__ float red[512];
  __shared__ float att_s[Lsz];
  __shared__ float dw_s[Tsz];
  __shared__ float stat[2];

  float local = 0.0f;
  for (int l = tid; l < Lsz; l += 512) {
    float ee = expe[b * Lsz + l];
    float ts = tsum[b * Lsz + l];
    float denom = (i == 0) ? 1.0f : fmaxf(ts, 1e-10f);
    float tmp = ee / denom;
    att_s[l] = tmp; local += tmp;
    tsum[b * Lsz + l] = ts + ee;
  }
  red[tid] = local; __syncthreads();
  for (int s = 256; s > 0; s >>= 1) { if (tid < s) red[tid] += red[tid + s]; __syncthreads(); }
  float S = red[0] + 1e-10f;
  for (int l = tid; l < Lsz; l += 512) { att_s[l] /= S; att[b * Lsz + l] = att_s[l]; }

  if (tid < Tsz) dw_s[tid] = (tid < i) ? dvals[b * Tsz + tid] : -1.0e9f;
  __syncthreads();
  if (tid == 0) {
    float mx = -1.0e30f;
    for (int t = 0; t < Tsz; ++t) mx = fmaxf(mx, dw_s[t]);
    float sm = 0.0f;
    for (int t = 0; t < Tsz; ++t) { dw_s[t] = __expf(dw_s[t] - mx); sm += dw_s[t]; }
    stat[0] = sm;
  }
  __syncthreads();
  float dsum = stat[0];

  // contexts + comb (tid == column h of H2)
  float ec = 0.0f, dc = 0.0f;
  for (int l = 0; l < Lsz; ++l) ec += att_s[l] * enc[((size_t)b * Lsz + l) * H2sz + tid];
  if (i > 0) {
    for (int t = 0; t < i; ++t) dc += (dw_s[t] / dsum) * pre_h[((size_t)b * Tsz + t) * H2sz + tid];
  }
  float h2v = pre_h[((size_t)b * Tsz + i) * H2sz + tid];
  comb[(size_t)b * 1536 + tid] = h2v;
  comb[(size_t)b * 1536 + H2sz + tid] = ec;
  comb[(size_t)b * 1536 + 2 * H2sz + tid] = dc;
  __syncthreads();

  // p = sigmoid(comb . Wp + bp)
  float pp = h2v * Wp[tid] + ec * Wp[H2sz + tid] + dc * Wp[2 * H2sz + tid];
  red[tid] = pp; __syncthreads();
  for (int s = 256; s > 0; s >>= 1) { if (tid < s) red[tid] += red[tid + s]; __syncthreads(); }
  if (tid == 0) pval[b] = sigf(red[0] + bp[0]);
}

// ---------------- vocab GEMM: logits[16][50000] = comb @ Wv + bv ----------------
// TDM stages 32x128 fp32 tiles of Wv into LDS (double-buffered, TENSORcnt-tracked),
// WMMA consumes B-operands from LDS while the next tile is in flight.
#define KT  32
#define NTB 128

#if HAVE_TDM
__device__ __forceinline__ void tdm_issue(const float* Wv, int n0b, int kt, unsigned lds_addr) {
  unsigned long long ga = (unsigned long long)(const void*)(Wv + (size_t)kt * Vsz + n0b);
  v4u g0;
  g0.x = 1u;                                                    // count = 1 valid descriptor
  g0.y = lds_addr;                                              // LDS byte address
  g0.z = (unsigned)(ga & 0xFFFFFFFFull);                        // global_addr[31:0]
  g0.w = (unsigned)((ga >> 32) & 0x01FFFFFFull) | (2u << 30);   // global_addr[56:32] | type=2
  unsigned td0 = (unsigned)(Vsz - n0b);                         // remaining cols (OOB clamp)
  unsigned td1 = (unsigned)(1536 - kt);                         // remaining rows
  v8i g1;
  g1[0] = (int)(2u << 16);                                      // data_size = 4B
  g1[1] = (int)((td0 & 0xFFFFu) << 16);                         // tensor_dim0 lo16 @ bits63:48
  g1[2] = (int)(((td0 >> 16) & 0xFFFFu) | ((td1 & 0xFFFFu) << 16));
  g1[3] = (int)(((td1 >> 16) & 0xFFFFu) | ((unsigned)NTB << 16)); // tile_dim0 = 128
  g1[4] = (int)(KT & 0xFFFF);                                   // tile_dim1 = 32, tile_dim2 = 0
  g1[5] = (int)Vsz;                                             // tensor_dim0_stride lo32
  g1[6] = 0;
  g1[7] = 0;
  v4i g2 = {0, 0, 0, 0};
  v4i g3 = {0, 0, 0, 0};
#if defined(__clang_major__) && (__clang_major__ >= 23)
  v8i g4 = {0, 0, 0, 0, 0, 0, 0, 0};
  __builtin_amdgcn_tensor_load_to_lds(g0, g1, g2, g3, g4, 0);
#else
  __builtin_amdgcn_tensor_load_to_lds(g0, g1, g2, g3, 0);
#endif
}
#endif

__global__ __launch_bounds__(256) void vocab_kernel(const float* __restrict__ comb,
                                                    const float* __restrict__ Wv,
                                                    const float* __restrict__ bvv,
                                                    float* __restrict__ logits) {
  const int tid = threadIdx.x;
  const int wave = tid >> 5, lane = tid & 31;
  const int m15 = lane & 15, kh = lane >> 4;
  const int n0b = blockIdx.x * NTB;
  const int n = n0b + wave * 16 + m15;
  const bool valid = (n0b + wave * 16 + 16) <= Vsz;
  const int nc = valid ? n : (Vsz - 1);

  v8f acc; for (int r = 0; r < 8; ++r) acc[r] = bvv[nc];

#if HAVE_TDM
  __shared__ float tile[2][KT * NTB];
  const unsigned lds0 = (unsigned)(size_t)(void*)&tile[0][0];
  const unsigned lds1 = (unsigned)(size_t)(void*)&tile[1][0];
  if (tid < 32) {                       // wave 0 drives the TDM
    tdm_issue(Wv, n0b, 0, lds0);
    __builtin_amdgcn_s_wait_tensorcnt(0);
  }
  __syncthreads();
  const int NCH = 1536 / KT;            // 48 chunks
  const int wcol = wave * 16 + m15;
  for (int c = 0; c < NCH; ++c) {
    const int kt = c * KT;
    if (tid < 32 && (c + 1) < NCH)
      tdm_issue(Wv, n0b, kt + KT, ((c + 1) & 1) ? lds1 : lds0);
    const float* buf = tile[c & 1];
    const float* arow = comb + (size_t)m15 * 1536 + kt;
#pragma unroll
    for (int k = 0; k < KT; k += 4) {
      int kk = k + 2 * kh;
      v2f a; a.x = arow[kk]; a.y = arow[kk + 1];
      v2f bm; bm.x = buf[kk * NTB + wcol]; bm.y = buf[(kk + 1) * NTB + wcol];
      acc = WMMA4(a, bm, acc);
    }
    if (tid < 32) __builtin_amdgcn_s_wait_tensorcnt(0);
    __syncthreads();
  }
#else
  const float* arow = comb + (size_t)m15 * 1536;
#pragma unroll 4
  for (int k = 0; k < 1536; k += 4) {
    int kk = k + 2 * kh;
    if ((k & 63) == 0 && k + 64 < 1536)
      __builtin_prefetch(Wv + (size_t)(kk + 64) * Vsz + nc, 0, 0);
    v2f a; a.x = arow[kk]; a.y = arow[kk + 1];
    v2f bm; bm.x = Wv[(size_t)kk * Vsz + nc]; bm.y = Wv[(size_t)(kk + 1) * Vsz + nc];
    acc = WMMA4(a, bm, acc);
  }
#endif
  if (valid)
    for (int r = 0; r < 8; ++r) logits[(size_t)(r + 8 * kh) * Vsz + n] = acc[r];
}

// ---------------- row max / sumexp over 50000 ----------------
__global__ __launch_bounds__(512) void softmax_stats_kernel(const float* __restrict__ logits,
                                                            float* __restrict__ rmax,
                                                            float* __restrict__ rsum) {
  const int b = blockIdx.x, tid = threadIdx.x;
  __shared__ float red[512];
  const float* row = logits + (size_t)b * Vsz;
  float mx = -1.0e30f;
  for (int v = tid; v < Vsz; v += 512) mx = fmaxf(mx, row[v]);
  red[tid] = mx; __syncthreads();
  for (int s = 256; s > 0; s >>= 1) { if (tid < s) red[tid] = fmaxf(red[tid], red[tid + s]); __syncthreads(); }
  mx = red[0]; __syncthreads();
  float sm = 0.0f;
  for (int v = tid; v < Vsz; v += 512) sm += __expf(row[v] - mx);
  red[tid] = sm; __syncthreads();
  for (int s = 256; s > 0; s >>= 1) { if (tid < s) red[tid] += red[tid + s]; __syncthreads(); }
  if (tid == 0) { rmax[b] = mx; rsum[b] = red[0]; }
}

// ---------------- final dist (vocab part), then ptr scatter, then argmax ----------------
__global__ __launch_bounds__(512) void final_dist_kernel(const float* __restrict__ logits,
                                                         const float* __restrict__ rmax,
                                                         const float* __restrict__ rsum,
                                                         const float* __restrict__ pval,
                                                         float* __restrict__ out, int i) {
  const int b = blockIdx.x;
  const int v = blockIdx.y * 512 + threadIdx.x;
  if (v >= VEXT) return;
  float val = 0.0f;
  if (v < Vsz)
    val = (1.0f - pval[b]) * __expf(logits[(size_t)b * Vsz + v] - rmax[b]) / rsum[b];
  out[((size_t)b * Tsz + i) * VEXT + v] = val;
}

__global__ __launch_bounds__(512) void scatter_ptr_kernel(const int* __restrict__ evx,
                                                          const float* __restrict__ att,
                                                          const float* __restrict__ pval,
                                                          float* __restrict__ out, int i) {
  const int b = blockIdx.x;
  float* row = out + ((size_t)b * Tsz + i) * VEXT;
  float p = pval[b];
  for (int l = threadIdx.x; l < Lsz; l += 512)
    atomicAdd(&row[evx[b * Lsz + l]], p * att[b * Lsz + l]);
}

__global__ __launch_bounds__(512) void argmax_kernel(const float* __restrict__ out,
                                                     int* __restrict__ dec_in, int i) {
  const int b = blockIdx.x, tid = threadIdx.x;
  __shared__ float sval[512];
  __shared__ int sidx[512];
  const float* row = out + ((size_t)b * Tsz + i) * VEXT;
  float best = -1.0e30f; int bi = 0;
  for (int v = tid; v < VEXT; v += 512) {
    float x = row[v];
    if (x > best) { best = x; bi = v; }
  }
  sval[tid] = best; sidx[tid] = bi; __syncthreads();
  for (int s = 256; s > 0; s >>= 1) {
    if (tid < s) {
      if (sval[tid + s] > sval[tid] ||
          (sval[tid + s] == sval[tid] && sidx[tid + s] < sidx[tid])) {
        sval[tid] = sval[tid + s]; sidx[tid] = sidx[tid + s];
      }
    }
    __syncthreads();
  }
  if (tid == 0) dec_in[b] = sidx[0];
}

// ---------------- host orchestration ----------------
extern "C" void kernel_launch(void* const* d_in, const int* in_sizes, int n_in,
                              void* d_out, int out_size, void* d_ws, size_t ws_size,
                              hipStream_t stream) {
  const int*   x         = (const int*)  d_in[0];
  const int*   x_len     = (const int*)  d_in[1];
  const int*   evx       = (const int*)  d_in[2];
  // d_in[3] = max_oov_len (constant 50, folded into VEXT)
  const float* embedding = (const float*)d_in[4];
  const float* Wx_f      = (const float*)d_in[5];
  const float* Wh_f      = (const float*)d_in[6];
  const float* b_f       = (const float*)d_in[7];
  const float* Wx_b      = (const float*)d_in[8];
  const float* Wh_b      = (const float*)d_in[9];
  const float* b_b       = (const float*)d_in[10];
  const float* Wx_d      = (const float*)d_in[11];
  const float* Wh_d      = (const float*)d_in[12];
  const float* b_d       = (const float*)d_in[13];
  const float* W_enc     = (const float*)d_in[14];
  const float* W_dec     = (const float*)d_in[15];
  const float* Wp        = (const float*)d_in[16];
  const float* bp        = (const float*)d_in[17];
  const float* Wv        = (const float*)d_in[18];
  const float* bv        = (const float*)d_in[19];
  float* out = (float*)d_out;

  // workspace partition (floats)
  float* f     = (float*)d_ws;
  float* emb   = f; f += (size_t)Bsz * Lsz * Esz;
  float* enc   = f; f += (size_t)Bsz * Lsz * H2sz;
  float* h0    = f; f += Bsz * H2sz;
  float* c0    = f; f += Bsz * H2sz;
  float* cping = f; f += Bsz * H2sz;
  float* cpong = f; f += Bsz * H2sz;
  float* pre_h = f; f += (size_t)Bsz * Tsz * H2sz;
  float* tsum  = f; f += Bsz * Lsz;
  float* expe  = f; f += Bsz * Lsz;
  float* att   = f; f += Bsz * Lsz;
  float* q     = f; f += Bsz * H2sz;
  float* qd    = f; f += Bsz * H2sz;
  float* dvals = f; f += Bsz * Tsz;
  float* comb  = f; f += Bsz * 1536;
  float* pval  = f; f += Bsz;
  float* logits= f; f += (size_t)Bsz * Vsz;
  float* rmax  = f; f += Bsz;
  float* rsum  = f; f += Bsz;
  int*   decin = (int*)f;

  init_kernel<<<1, 256, 0, stream>>>(tsum, decin);
  embed_kernel<<<Bsz * Lsz, 128, 0, stream>>>(x, x_len, embedding, emb);
  enc_lstm_kernel<<<2, 512, 0, stream>>>(emb, x_len, Wx_f, Wh_f, b_f, Wx_b, Wh_b, b_b,
                                         enc, h0, c0);

  for (int i = 0; i < Tsz; ++i) {
    const float* c_in = (i == 0) ? c0 : ((i & 1) ? cping : cpong);
    float* c_out = (i & 1) ? cpong : cping;
    dec_cell_kernel<<<32, 32, 0, stream>>>(embedding, decin, Wx_d, Wh_d, b_d,
                                           h0, c_in, pre_h, c_out, i);
    attn_q_kernel<<<64, 32, 0, stream>>>(pre_h, W_enc, W_dec, q, qd, i);
    attn_scores_kernel<<<Bsz, 512, 0, stream>>>(q, qd, enc, pre_h, x_len, expe, dvals);
    attn_finalize_kernel<<<Bsz, 512, 0, stream>>>(expe, tsum, enc, dvals, pre_h,
                                                  Wp, bp, att, comb, pval, i);
    vocab_kernel<<<(Vsz + NTB - 1) / NTB, 256, 0, stream>>>(comb, Wv, bv, logits);
    softmax_stats_kernel<<<Bsz, 512, 0, stream>>>(logits, rmax, rsum);
    final_dist_kernel<<<dim3(Bsz, (VEXT + 511) / 512), 512, 0, stream>>>(logits, rmax, rsum,
                                                                         pval, out, i);
    scatter_ptr_kernel<<<Bsz, 512, 0, stream>>>(evx, att, pval, out, i);
    argmax_kernel<<<Bsz, 512, 0, stream>>>(out, decin, i);
  }
}